// HoroPCA_38156489458121
// MI455X (gfx1250) — compile-verified
//
#include <hip/hip_runtime.h>
#include <math.h>

typedef float v2f __attribute__((ext_vector_type(2)));
typedef float v8f __attribute__((ext_vector_type(8)));

#define N_PTS 8192
#define DIM   64
#define DP1   65
#define KPAD  68
#define NCOMP 8

// workspace byte offsets
#define WS_ACC  0        // double accumulator
#define WS_UT   64       // 72 floats (unit tangent, padded)
#define WS_GINV 352      // 64 floats (8x8 gram inverse)
#define WS_QH   608      // 8*KPAD floats (lifted ideal points, padded)
#define WS_XP   4096     // 8192*KPAD floats (projected points, padded)

__device__ __forceinline__ float waveReduce(float x) {
#pragma unroll
  for (int off = 16; off >= 1; off >>= 1) x += __shfl_xor(x, off, 32);
  return x;
}

// ---------------- Phase 0: orthonormalize + gram inverse + tangent -----------
__global__ __launch_bounds__(32)
void horo_setup(const float* __restrict__ Qp, float* __restrict__ wsf,
                double* __restrict__ acc) {
  __shared__ float V[NCOMP][DIM];
  __shared__ float G[NCOMP * NCOMP];
  __shared__ float GJ[NCOMP][2 * NCOMP];
  __shared__ float coef[NCOMP];
  const int lane = threadIdx.x;

  // each lane owns dims (lane, lane+32) of every row
  float v0[NCOMP], v1[NCOMP];
#pragma unroll
  for (int i = 0; i < NCOMP; i++) {
    v0[i] = Qp[i * DIM + lane];
    v1[i] = Qp[i * DIM + lane + 32];
  }
  // modified Gram-Schmidt over the 8 rows
#pragma unroll
  for (int i = 0; i < NCOMP; i++) {
#pragma unroll
    for (int j = 0; j < NCOMP; j++) {
      if (j < i) {
        float r = waveReduce(v0[j] * v0[i] + v1[j] * v1[i]);
        v0[i] -= r * v0[j];
        v1[i] -= r * v1[j];
      }
    }
    float n2 = waveReduce(v0[i] * v0[i] + v1[i] * v1[i]);
    float inv = rsqrtf(fmaxf(n2, 1e-30f));
    v0[i] *= inv;
    v1[i] *= inv;
  }
#pragma unroll
  for (int i = 0; i < NCOMP; i++) {
    V[i][lane] = v0[i];
    V[i][lane + 32] = v1[i];
  }
  __syncthreads();

  // Minkowski gram of lifted rows Q=[1,v]: g_ij = -1 + v_i.v_j
  for (int e = lane; e < NCOMP * NCOMP; e += 32) {
    int i = e >> 3, j = e & 7;
    float s = 0.f;
    for (int d = 0; d < DIM; d++) s += V[i][d] * V[j][d];
    G[e] = s - 1.0f;
  }
  __syncthreads();

  if (lane == 0) {
    // Gauss-Jordan inverse with partial pivoting (8x8)
    for (int i = 0; i < NCOMP; i++)
      for (int j = 0; j < NCOMP; j++) {
        GJ[i][j] = G[i * 8 + j];
        GJ[i][8 + j] = (i == j) ? 1.f : 0.f;
      }
    for (int col = 0; col < NCOMP; col++) {
      int p = col;
      float best = fabsf(GJ[col][col]);
      for (int r = col + 1; r < NCOMP; r++) {
        float a = fabsf(GJ[r][col]);
        if (a > best) { best = a; p = r; }
      }
      if (p != col)
        for (int j = 0; j < 16; j++) {
          float t = GJ[col][j]; GJ[col][j] = GJ[p][j]; GJ[p][j] = t;
        }
      float inv = 1.0f / GJ[col][col];
      for (int j = 0; j < 16; j++) GJ[col][j] *= inv;
      for (int r = 0; r < NCOMP; r++)
        if (r != col) {
          float f = GJ[r][col];
          for (int j = 0; j < 16; j++) GJ[r][j] -= f * GJ[col][j];
        }
    }
  }
  __syncthreads();

  // write gram inverse
  float* ginv = wsf + WS_GINV / 4;
  for (int e = lane; e < 64; e += 32) ginv[e] = GJ[e >> 3][8 + (e & 7)];

  // write lifted Q rows, stride KPAD, zero-padded
  float* Qh = wsf + WS_QH / 4;
  for (int e = lane; e < NCOMP * KPAD; e += 32) {
    int k = e / KPAD, t = e % KPAD;
    float val = (t == 0) ? 1.0f : ((t <= DIM) ? V[k][t - 1] : 0.0f);
    Qh[e] = val;
  }

  // tangent at origin: <origin,Q_k>_L = -1  ->  coef_k = -sum_j ginv[j][k]
  if (lane < NCOMP) {
    float s = 0.f;
    for (int j = 0; j < NCOMP; j++) s += GJ[j][8 + lane];
    coef[lane] = -s;
  }
  __syncthreads();
  float csum = 0.f;
#pragma unroll
  for (int k = 0; k < NCOMP; k++) csum += coef[k];
  float t_time = 1.0f - csum;             // tangent0 time component
  float s0 = 0.f, s1 = 0.f;
#pragma unroll
  for (int k = 0; k < NCOMP; k++) {
    s0 += coef[k] * V[k][lane];
    s1 += coef[k] * V[k][lane + 32];
  }
  float tsp0 = -s0, tsp1 = -s1;           // spatial components
  float tin = waveReduce(tsp0 * tsp0 + tsp1 * tsp1) - t_time * t_time;
  float inv = rsqrtf(fmaxf(tin, 1e-7f));
  float* ut = wsf + WS_UT / 4;
  ut[1 + lane] = tsp0 * inv;
  ut[1 + lane + 32] = tsp1 * inv;
  if (lane == 0) {
    ut[0] = t_time * inv;
    for (int t = DP1; t < 72; t++) ut[t] = 0.f;
    acc[0] = 0.0;                         // zero global accumulator each call
  }
}

// ---------------- Phase 1: per-point horospherical projection + expmap -------
__global__ __launch_bounds__(256)
void horo_project(const float* __restrict__ x_data, const float* __restrict__ wsf,
                  float* __restrict__ xp) {
  __shared__ float Qh[NCOMP * KPAD];
  __shared__ float GI[64];
  __shared__ float UT[DP1];
  const int tid = threadIdx.x;
  const float* wQh = wsf + WS_QH / 4;
  const float* wGI = wsf + WS_GINV / 4;
  const float* wUT = wsf + WS_UT / 4;
  for (int e = tid; e < NCOMP * KPAD; e += 256) Qh[e] = wQh[e];
  for (int e = tid; e < 64; e += 256) GI[e] = wGI[e];
  for (int e = tid; e < DP1; e += 256) UT[e] = wUT[e];
  __syncthreads();

  const int i = blockIdx.x * 256 + tid;
  if (i >= N_PTS) return;

  float x[DP1];
#pragma unroll
  for (int t = 0; t < DP1; t++) x[t] = x_data[(long)i * DP1 + t];

  // Minkowski inner products with lifted components
  float xq[NCOMP];
#pragma unroll
  for (int k = 0; k < NCOMP; k++) {
    float a = -x[0] * Qh[k * KPAD];
#pragma unroll
    for (int t = 1; t < DP1; t++) a += x[t] * Qh[k * KPAD + t];
    xq[k] = a;
  }
  float cf[NCOMP];
#pragma unroll
  for (int k = 0; k < NCOMP; k++) {
    float s = 0.f;
#pragma unroll
    for (int j = 0; j < NCOMP; j++) s += xq[j] * GI[j * 8 + k];
    cf[k] = s;
  }
  // mink_proj
  float p[DP1];
#pragma unroll
  for (int t = 0; t < DP1; t++) {
    float s = 0.f;
#pragma unroll
    for (int k = 0; k < NCOMP; k++) s += cf[k] * Qh[k * KPAD + t];
    p[t] = s;
  }
  float pin = -p[0] * p[0];
#pragma unroll
  for (int t = 1; t < DP1; t++) pin += p[t] * p[t];
  float sinv = rsqrtf(fmaxf(-pin, 1e-7f));
#pragma unroll
  for (int t = 0; t < DP1; t++) p[t] *= sinv;       // spine

  float spi = -x[0] * p[0];
#pragma unroll
  for (int t = 1; t < DP1; t++) spi += x[t] * p[t];
  float z = fmaxf(-spi, 1.0f + 1e-7f);
  float sd = acoshf(z);                              // geodesic distance to spine

  float uu = -UT[0] * UT[0];
#pragma unroll
  for (int t = 1; t < DP1; t++) uu += UT[t] * UT[t];
  float vn = sqrtf(fmaxf(sd * sd * uu, 1e-7f));
  float ch = coshf(vn);
  float sh = sinhf(vn) / vn;

#pragma unroll
  for (int t = 0; t < DP1; t++)
    xp[(long)i * KPAD + t] = ch * p[t] + sh * sd * UT[t];
#pragma unroll
  for (int t = DP1; t < KPAD; t++) xp[(long)i * KPAD + t] = 0.f;
}

// ---------------- Phase 2: pairwise Minkowski Gram via fp32 WMMA -------------
#define BM 128
#define BN 64

__global__ __launch_bounds__(256)
void horo_pairwise(const float* __restrict__ xp, double* __restrict__ acc) {
  __shared__ float A[BM * KPAD];   // 34816 B
  __shared__ float B[BN * KPAD];   // 17408 B
  __shared__ float red[256];
  const int tid = threadIdx.x;
  const int wave = tid >> 5;
  const int lane = tid & 31;
  const int ln16 = lane & 15;
  const int hi = lane >> 4;

  const float* gA = xp + (long)blockIdx.x * BM * KPAD;
  const float* gB = xp + (long)blockIdx.y * BN * KPAD;
  for (int e = tid; e < BM * KPAD; e += 256) A[e] = gA[e];
  for (int e = tid; e < BN * KPAD; e += 256) {
    float v = gB[e];
    if (e % KPAD == 0) v = -v;    // fold Minkowski sign into B's time coord
    B[e] = v;
  }
  __syncthreads();

  const int arow = wave * 16 + ln16;     // this wave's A fragment row
  const int koff = hi * 2;               // lanes 16-31 carry K+2 (ISA 16x4 layout)

  v8f zero = {0.f, 0.f, 0.f, 0.f, 0.f, 0.f, 0.f, 0.f};
  v8f c[4];
#pragma unroll
  for (int ct = 0; ct < 4; ct++) c[ct] = zero;

#pragma unroll
  for (int kt = 0; kt < KPAD / 4; kt++) {
    const int k = 4 * kt + koff;
    v2f a = *(const v2f*)&A[arow * KPAD + k];
#pragma unroll
    for (int ct = 0; ct < 4; ct++) {
      v2f b = *(const v2f*)&B[(ct * 16 + ln16) * KPAD + k];
      c[ct] = __builtin_amdgcn_wmma_f32_16x16x4_f32(
          false, a, false, b, (short)0, c[ct], false, false);
    }
  }

  // fused arccosh^2 reduction over accumulator fragments (layout-agnostic)
  float part = 0.f;
#pragma unroll
  for (int ct = 0; ct < 4; ct++) {
#pragma unroll
    for (int v = 0; v < 8; v++) {
      float m = c[ct][v];
      float zz = fmaxf(-m, 1.0f + 1e-7f);
      float d = acoshf(zz);
      part += d * d;
    }
  }
  red[tid] = part;
  __syncthreads();
  for (int s = 128; s > 0; s >>= 1) {
    if (tid < s) red[tid] += red[tid + s];
    __syncthreads();
  }
  if (tid == 0) atomicAdd(acc, (double)red[0]);
}

// ---------------- Phase 3: finalize ------------------------------------------
__global__ void horo_finalize(const double* __restrict__ acc, float* __restrict__ out) {
  if (threadIdx.x == 0 && blockIdx.x == 0) {
    const double nn = (double)N_PTS * (double)N_PTS;
    out[0] = (float)(-(acc[0] / nn));
  }
}

extern "C" void kernel_launch(void* const* d_in, const int* in_sizes, int n_in,
                              void* d_out, int out_size, void* d_ws, size_t ws_size,
                              hipStream_t stream) {
  const float* x_data = (const float*)d_in[0];   // (8192, 65) f32
  const float* Q_param = (const float*)d_in[1];  // (8, 64)    f32
  float* wsf = (float*)d_ws;
  double* acc = (double*)d_ws;                   // offset WS_ACC == 0
  float* xp = (float*)((char*)d_ws + WS_XP);     // (8192, 68) f32

  horo_setup<<<1, 32, 0, stream>>>(Q_param, wsf, acc);
  horo_project<<<N_PTS / 256, 256, 0, stream>>>(x_data, wsf, xp);
  dim3 grid(N_PTS / BM, N_PTS / BN);
  horo_pairwise<<<grid, 256, 0, stream>>>(xp, acc);
  horo_finalize<<<1, 1, 0, stream>>>(acc, (float*)d_out);
}